// GCN_12841952215814
// MI455X (gfx1250) — compile-verified
//
#include <hip/hip_runtime.h>
#include <math.h>

#define N_NODES   100000
#define N_EDGES   1600000
#define HID       128
#define N_CLASSES 16
#define N_GRAPHS  512

typedef float v2f __attribute__((ext_vector_type(2)));
typedef float v8f __attribute__((ext_vector_type(8)));

// Hardware f32 atomic add (no-return form, STOREcnt-tracked). Inline asm so we
// are guaranteed global_atomic_add_f32 and never a CAS loop.
__device__ __forceinline__ void gatomic_add_f32(float* p, float v) {
    asm volatile("global_atomic_add_f32 %0, %1, off" :: "v"(p), "v"(v) : "memory");
}

__global__ void zero_kernel(float* __restrict__ p, int n) {
    int i = blockIdx.x * blockDim.x + threadIdx.x;
    int stride = gridDim.x * blockDim.x;
    for (; i < n; i += stride) p[i] = 0.0f;
}

__global__ void degree_kernel(const int* __restrict__ src, const int* __restrict__ dst,
                              float* deg_out, float* deg_in) {
    int e = blockIdx.x * blockDim.x + threadIdx.x;
    if (e >= N_EDGES) return;
    gatomic_add_f32(deg_out + src[e], 1.0f);
    gatomic_add_f32(deg_in  + dst[e], 1.0f);
}

__global__ void rsqrt_clip_kernel(float* p, int n) {
    int i = blockIdx.x * blockDim.x + threadIdx.x;
    if (i < n) {
        float d = p[i];
        d = d < 1.0f ? 1.0f : d;
        p[i] = rsqrtf(d);
    }
}

// One wave (32 lanes) per edge; lane l owns channels [4l, 4l+4).
// agg[dst] += h[src] * norm_src[src]
__global__ void scatter_kernel(const float* __restrict__ h, const float* __restrict__ norm_src,
                               const int* __restrict__ src, const int* __restrict__ dst,
                               float* agg) {
    int wave = (blockIdx.x * blockDim.x + threadIdx.x) >> 5;
    int lane = threadIdx.x & 31;
    if (wave >= N_EDGES) return;
    int s = src[wave], d = dst[wave];
    float ns = norm_src[s];
    float4 v = ((const float4*)(h + (size_t)s * HID))[lane];
    float* out = agg + (size_t)d * HID + lane * 4;
    gatomic_add_f32(out + 0, v.x * ns);
    gatomic_add_f32(out + 1, v.y * ns);
    gatomic_add_f32(out + 2, v.z * ns);
    gatomic_add_f32(out + 3, v.w * ns);
}

// In-place: agg <- relu((agg * norm_dst) @ W + b)
// Wave computes a 16x128 strip with V_WMMA_F32_16X16X4_F32.
// A-frag layout (ISA 7.12.2, 32-bit A 16x4): lanes 0-15 hold K={0,1}, lanes 16-31 K={2,3}.
// B-frag (4x16): lanes 0-15 N=lane K={0,1}; lanes 16-31 N=lane-16 K={2,3}.
// C/D: VGPR r -> M = r (+8 for upper half-lanes), N = lane & 15.
__global__ void __launch_bounds__(256)
gemm_bias_relu_kernel(float* __restrict__ agg, const float* __restrict__ norm_dst,
                      const float* __restrict__ W, const float* __restrict__ b) {
    __shared__ float Wt[HID * HID];            // W transposed: Wt[n*128 + k]
    for (int i = threadIdx.x; i < HID * HID; i += blockDim.x) {
        int k = i >> 7, n = i & 127;
        Wt[n * HID + k] = W[i];
    }
    __syncthreads();

    int wave = threadIdx.x >> 5;
    int lane = threadIdx.x & 31;
    int half = lane >> 4;                      // 0: K lo pair, 1: K hi pair
    int m16  = lane & 15;
    int rowBase = blockIdx.x * 128 + wave * 16;
    if (rowBase >= N_NODES) return;            // N_NODES % 16 == 0: tiles are full

    int row = rowBase + m16;
    float scale = norm_dst[row];

    // Preload entire 16x128 A strip into registers (in-place safety).
    float a[64];
    const float* arow = agg + (size_t)row * HID + 2 * half;
    #pragma unroll
    for (int ks = 0; ks < 32; ++ks) {          // k0 = 4*ks + 2*half
        float2 av = *(const float2*)(arow + 4 * ks);
        a[2 * ks]     = av.x * scale;
        a[2 * ks + 1] = av.y * scale;
    }

    #pragma unroll
    for (int nt = 0; nt < 8; ++nt) {
        int n = nt * 16 + m16;
        v8f acc = {0.f, 0.f, 0.f, 0.f, 0.f, 0.f, 0.f, 0.f};
        const float* wcol = &Wt[n * HID + 2 * half];
        #pragma unroll
        for (int ks = 0; ks < 32; ++ks) {
            v2f A; A[0] = a[2 * ks]; A[1] = a[2 * ks + 1];
            float2 bw = *(const float2*)(wcol + 4 * ks);
            v2f B; B[0] = bw.x; B[1] = bw.y;
            acc = __builtin_amdgcn_wmma_f32_16x16x4_f32(
                false, A, false, B, (short)0, acc, false, false);
        }
        float bias = b[n];
        #pragma unroll
        for (int r = 0; r < 8; ++r) {
            int orow = rowBase + (half ? r + 8 : r);
            float v = acc[r] + bias;
            agg[(size_t)orow * HID + n] = v > 0.0f ? v : 0.0f;
        }
    }
}

// w[i] = sigmoid(dot(h[i], Wv) + bv); one wave per node.
__global__ void attn_kernel(const float* __restrict__ h, const float* __restrict__ Wv,
                            const float* __restrict__ bv, float* w) {
    int node = (blockIdx.x * blockDim.x + threadIdx.x) >> 5;
    int lane = threadIdx.x & 31;
    if (node >= N_NODES) return;
    float4 hv = ((const float4*)(h + (size_t)node * HID))[lane];
    float4 wv = ((const float4*)Wv)[lane];
    float p = hv.x * wv.x + hv.y * wv.y + hv.z * wv.z + hv.w * wv.w;
    #pragma unroll
    for (int off = 16; off; off >>= 1) p += __shfl_xor(p, off, 32);
    if (lane == 0) w[node] = 1.0f / (1.0f + expf(-(p + bv[0])));
}

// num[gid] += h*w ; den[gid] += w  (wave per node)
__global__ void readout_kernel(const float* __restrict__ h, const float* __restrict__ w,
                               const int* __restrict__ gid, float* num, float* den) {
    int node = (blockIdx.x * blockDim.x + threadIdx.x) >> 5;
    int lane = threadIdx.x & 31;
    if (node >= N_NODES) return;
    int g = gid[node];
    float wt = w[node];
    float4 hv = ((const float4*)(h + (size_t)node * HID))[lane];
    float* np = num + (size_t)g * HID + lane * 4;
    gatomic_add_f32(np + 0, hv.x * wt);
    gatomic_add_f32(np + 1, hv.y * wt);
    gatomic_add_f32(np + 2, hv.z * wt);
    gatomic_add_f32(np + 3, hv.w * wt);
    if (lane == 0) gatomic_add_f32(den + g, wt);
}

// out[g,c] = (den>0 ? num[g]/den : 0) @ Wc + bc   (512x16, trivial)
__global__ void final_kernel(const float* __restrict__ num, const float* __restrict__ den,
                             const float* __restrict__ Wc, const float* __restrict__ bc,
                             float* out) {
    int i = blockIdx.x * blockDim.x + threadIdx.x;
    if (i >= N_GRAPHS * N_CLASSES) return;
    int g = i / N_CLASSES, c = i % N_CLASSES;
    float d = den[g];
    float acc = 0.0f;
    if (d > 0.0f) {
        float inv = 1.0f / d;
        for (int k = 0; k < HID; ++k)
            acc += num[(size_t)g * HID + k] * inv * Wc[k * N_CLASSES + c];
    }
    out[i] = acc + bc[c];
}

extern "C" void kernel_launch(void* const* d_in, const int* in_sizes, int n_in,
                              void* d_out, int out_size, void* d_ws, size_t ws_size,
                              hipStream_t stream) {
    (void)in_sizes; (void)n_in; (void)out_size; (void)ws_size;
    const float* x   = (const float*)d_in[0];
    const int*   src = (const int*)d_in[1];
    const int*   dst = (const int*)d_in[2];
    const int*   gid = (const int*)d_in[3];
    const float* W[4]  = {(const float*)d_in[4], (const float*)d_in[6],
                          (const float*)d_in[8], (const float*)d_in[10]};
    const float* bb[4] = {(const float*)d_in[5], (const float*)d_in[7],
                          (const float*)d_in[9], (const float*)d_in[11]};
    const float* Wv = (const float*)d_in[12];
    const float* bv = (const float*)d_in[13];
    const float* Wc = (const float*)d_in[14];
    const float* bc = (const float*)d_in[15];
    float* out = (float*)d_out;

    // Workspace layout (floats): [deg_out|deg_in](2N) | buf0(N*128) | buf1(N*128)
    //                            | w(N) | num(G*128) | den(G)   ~= 104 MB
    float* ws       = (float*)d_ws;
    float* deg      = ws;                  // becomes norms in place
    float* norm_src = deg;
    float* norm_dst = deg + N_NODES;
    float* buf0     = deg + 2 * (size_t)N_NODES;
    float* buf1     = buf0 + (size_t)N_NODES * HID;
    float* wgt      = buf1 + (size_t)N_NODES * HID;
    float* num      = wgt + N_NODES;
    float* den      = num + (size_t)N_GRAPHS * HID;

    const int B = 256;

    zero_kernel<<<1024, B, 0, stream>>>(deg, 2 * N_NODES);
    degree_kernel<<<(N_EDGES + B - 1) / B, B, 0, stream>>>(src, dst, deg, deg + N_NODES);
    rsqrt_clip_kernel<<<(2 * N_NODES + B - 1) / B, B, 0, stream>>>(deg, 2 * N_NODES);

    const int scatterGrid = N_EDGES / 8;          // 8 waves (edges) per 256-thr block
    const int gemmGrid    = (N_NODES + 127) / 128;

    const float* hin = x;
    float* bufs[2] = {buf0, buf1};
    for (int layer = 0; layer < 4; ++layer) {
        float* tgt = bufs[layer & 1];
        zero_kernel<<<4096, B, 0, stream>>>(tgt, N_NODES * HID);
        scatter_kernel<<<scatterGrid, B, 0, stream>>>(hin, norm_src, src, dst, tgt);
        gemm_bias_relu_kernel<<<gemmGrid, B, 0, stream>>>(tgt, norm_dst, W[layer], bb[layer]);
        hin = tgt;
    }

    attn_kernel<<<N_NODES / 8, B, 0, stream>>>(hin, Wv, bv, wgt);
    zero_kernel<<<64, B, 0, stream>>>(num, N_GRAPHS * HID + N_GRAPHS);
    readout_kernel<<<N_NODES / 8, B, 0, stream>>>(hin, wgt, gid, num, den);
    final_kernel<<<(N_GRAPHS * N_CLASSES + B - 1) / B, B, 0, stream>>>(num, den, Wc, bc, out);
}